// SamplePDF_66838281060628
// MI455X (gfx1250) — compile-verified
//
#include <hip/hip_runtime.h>
#include <hip/hip_bf16.h>
#include <stddef.h>

// ---------------------------------------------------------------------------
// sample_pdf (NeRF inverse-CDF importance sampling) for MI455X / gfx1250.
// Memory-bound (~402 MB @ 23.3 TB/s => ~17us floor). Strategy:
//   * 16 rays per 256-thread block (8 wave32s)
//   * async global->LDS staging of weights/bins (ASYNCcnt path)
//   * in-place epsilon pass so LDS holds (w+1e-5, pad=0) -> branchless WMMA/scan
//   * per-ray totals via V_WMMA_F32_16X16X4_F32 with an all-ones B
//     (layout-safe: documented A 16x4 f32 and D 16x16 f32 layouts only)
//   * two-level LDS scan for the 64-wide CDF, then per-sample binary search
// ---------------------------------------------------------------------------

typedef __attribute__((ext_vector_type(2))) float v2f;
typedef __attribute__((ext_vector_type(8))) float v8f;
typedef int i32x4 __attribute__((vector_size(16)));  // matches builtin's param

#define RAYS_PER_BLOCK 16
#define NBINS 64   // bin edges per ray == cdf length
#define NW    63   // weights per ray
#define NIMP  128  // fine samples per ray
#define WS    68   // padded LDS row stride (floats): 272B rows -> 16B aligned,
                   // and bank = (4r + c) mod 64 -> conflict-free column gathers
#define STS   17   // segment-total row stride (17 invertible mod 64)

#if defined(__has_builtin)
#  if __has_builtin(__builtin_amdgcn_global_load_async_to_lds_b32) && \
      __has_builtin(__builtin_amdgcn_global_load_async_to_lds_b128)
#    define USE_ASYNC_LDS 1
#  endif
#  if __has_builtin(__builtin_amdgcn_wmma_f32_16x16x4_f32)
#    define USE_WMMA_F32 1
#  endif
#endif

// Typed address-space casts matching the builtin prototypes:
//   b32 : (as1 int*,   as3 int*,   imm offset, imm cpol)
//   b128: (as1 i32x4*, as3 i32x4*, imm offset, imm cpol)
#define GPTR32(p)  ((__attribute__((address_space(1))) int*)(p))
#define LPTR32(p)  ((__attribute__((address_space(3))) int*)(p))
#define GPTR128(p) ((__attribute__((address_space(1))) i32x4*)(p))
#define LPTR128(p) ((__attribute__((address_space(3))) i32x4*)(p))

__device__ __forceinline__ void wait_async0() {
#if defined(USE_ASYNC_LDS)
#  if __has_builtin(__builtin_amdgcn_s_wait_asynccnt)
  __builtin_amdgcn_s_wait_asynccnt(0);
#  else
  asm volatile("s_wait_asynccnt 0" ::: "memory");
#  endif
#endif
}

__global__ __launch_bounds__(256) void SamplePDF_kernel(
    const float* __restrict__ bins, const float* __restrict__ weights,
    const float* __restrict__ u, float* __restrict__ out, int n_rays) {
  __shared__ float w_s[RAYS_PER_BLOCK * WS];     // w+1e-5 after Phase B; col63=0
  __shared__ float bins_s[RAYS_PER_BLOCK * WS];  // bin edges
  __shared__ float cdf_s[RAYS_PER_BLOCK * WS];   // normalized cdf, 64 entries
  __shared__ float segtot_s[RAYS_PER_BLOCK * STS];
  __shared__ float tot_s[RAYS_PER_BLOCK];

  const int tid = threadIdx.x;
  const int ray0 = blockIdx.x * RAYS_PER_BLOCK;
  if (ray0 >= n_rays) return;

  const float* wg = weights + (size_t)ray0 * NW;
  const float* bg = bins + (size_t)ray0 * NBINS;

  // ---- Phase A: stage weights + bins into LDS (async-to-LDS path) ----
#if defined(USE_ASYNC_LDS)
#pragma unroll
  for (int kk = 0; kk < 4; ++kk) {
    int idx = kk * 256 + tid;          // 1024 slots = 16 rays x 64 cols
    int r = idx >> 6, c = idx & 63;
    if (c < NW) {
      __builtin_amdgcn_global_load_async_to_lds_b32(
          GPTR32(wg + r * NW + c), LPTR32(&w_s[r * WS + c]), 0, 0);
    } else {
      w_s[r * WS + c] = 0.0f;          // K-pad column (never aliases async dst)
    }
  }
  {
    int r = tid >> 4, cg = tid & 15;   // 16B-aligned b128 per thread
    __builtin_amdgcn_global_load_async_to_lds_b128(
        GPTR128(bg + r * NBINS + cg * 4), LPTR128(&bins_s[r * WS + cg * 4]),
        0, 0);
  }
  wait_async0();
#else
#pragma unroll
  for (int kk = 0; kk < 4; ++kk) {
    int idx = kk * 256 + tid;
    int r = idx >> 6, c = idx & 63;
    w_s[r * WS + c] = (c < NW) ? wg[r * NW + c] : 0.0f;
  }
  {
    int r = tid >> 4, cg = tid & 15;
    const float4 v = *(const float4*)(bg + r * NBINS + cg * 4);
    *(float4*)(&bins_s[r * WS + cg * 4]) = v;
  }
#endif
  __syncthreads();

  // ---- Phase B: in-place epsilon: w_s = w+1e-5 (cols<63), pad stays 0 ----
  // Keeps every later consumer (WMMA A-fragments, scan) branch-free with
  // uniform EXEC, instead of lane-divergent guards inside the WMMA loop.
#pragma unroll
  for (int kk = 0; kk < 4; ++kk) {
    int idx = kk * 256 + tid;
    int r = idx >> 6, c = idx & 63;
    if (c < NW) w_s[r * WS + c] += 1e-5f;
  }
  __syncthreads();

  // ---- Phase C: per-ray totals = rowsum(w_s) via f32 WMMA ----
  // D = sum_kb A_kb(16x4) x Ones(4x16): all-ones B is layout-invariant, so
  // only the documented A(16x4 f32) and D(16x16 f32) layouts are relied on.
#if defined(USE_WMMA_F32)
  if (tid < 32) {  // wave-uniform branch: EXEC all ones inside (WMMA req.)
    const int lane = tid;
    const int row = lane & 15;
    const int dsel = (lane >> 4) << 1;  // lanes 0-15: K+{0,1}; 16-31: K+{2,3}
    v2f bones;
    bones.x = 1.0f;
    bones.y = 1.0f;
    v8f acc = {};
#pragma unroll
    for (int kb = 0; kb < 16; ++kb) {
      const v2f a = *(const v2f*)(&w_s[row * WS + 4 * kb + dsel]);  // ds_load_b64
      acc = __builtin_amdgcn_wmma_f32_16x16x4_f32(
          false, a, false, bones, (short)0, acc, false, false);
    }
    // D layout: lane 0 col N=0 holds rows 0..7 in acc[0..7]; lane 16 rows 8..15
    if (lane == 0) {
#pragma unroll
      for (int q = 0; q < 8; ++q) tot_s[q] = acc[q];
    }
    if (lane == 16) {
#pragma unroll
      for (int q = 0; q < 8; ++q) tot_s[8 + q] = acc[q];
    }
  }
#endif

  // ---- Phase D: 64-wide inclusive scan per ray (two-level, in LDS) ----
  const int r = tid >> 4;  // ray within block
  const int s = tid & 15;  // segment of 4
  const float4 wv = *(const float4*)(&w_s[r * WS + 4 * s]);
  const float p0 = wv.x, p1 = p0 + wv.y, p2 = p1 + wv.z, p3 = p2 + wv.w;
  segtot_s[r * STS + s] = p3;
  __syncthreads();

  if (tid < RAYS_PER_BLOCK) {  // serial exclusive scan of 16 segment totals
    const int rr = tid;
    float run = 0.0f;
#pragma unroll
    for (int ss = 0; ss < 16; ++ss) {
      const float t = segtot_s[rr * STS + ss];
      segtot_s[rr * STS + ss] = run;
      run += t;
    }
#if !defined(USE_WMMA_F32)
    tot_s[rr] = run;  // fallback total if WMMA builtin unavailable
#endif
  }
  __syncthreads();

  {
    const float off = segtot_s[r * STS + s];
    const float inv = 1.0f / tot_s[r];
    const int base = 4 * s + 1;  // inclusive[j] -> cdf[j+1]
    if (s == 0) cdf_s[r * WS + 0] = 0.0f;
    cdf_s[r * WS + base + 0] = (off + p0) * inv;
    cdf_s[r * WS + base + 1] = (off + p1) * inv;
    cdf_s[r * WS + base + 2] = (off + p2) * inv;
    if (s < 15) cdf_s[r * WS + base + 3] = (off + p3) * inv;
  }
  __syncthreads();

  // ---- Phase E: 2048 samples; binary search + interp, coalesced IO ----
  const float* ub = u + (size_t)ray0 * NIMP;
  float* ob = out + (size_t)ray0 * NIMP;
#pragma unroll
  for (int k = 0; k < 8; ++k) {
    const int lin = k * 256 + tid;   // wave spans a single ray's cdf row
    const int rl = lin >> 7;
    const float uu = ub[lin];
    const float* crow = &cdf_s[rl * WS];
    const float* brow = &bins_s[rl * WS];
    int lo = 0, hi = NBINS;          // searchsorted(side='left')
#pragma unroll
    for (int it = 0; it < 6; ++it) {
      const int mid = (lo + hi) >> 1;
      const bool pred = crow[mid] < uu;
      lo = pred ? mid + 1 : lo;
      hi = pred ? hi : mid;
    }
    int lower = lo - 1; if (lower < 0) lower = 0;
    int upper = lo;     if (upper > NBINS - 1) upper = NBINS - 1;
    const float clo = crow[lower], chi = crow[upper];
    const float blo = brow[lower], bhi = brow[upper];
    float denom = chi - clo;
    denom = (denom < 1e-5f) ? 1.0f : denom;
    const float t = (uu - clo) / denom;
    ob[lin] = blo + t * (bhi - blo);
  }
}

extern "C" void kernel_launch(void* const* d_in, const int* in_sizes, int n_in,
                              void* d_out, int out_size, void* d_ws,
                              size_t ws_size, hipStream_t stream) {
  const float* bins = (const float*)d_in[0];     // [N, 64]
  const float* weights = (const float*)d_in[1];  // [N, 63]
  const float* u = (const float*)d_in[2];        // [N, 128]
  float* out = (float*)d_out;                    // [N, 128]
  const int n_rays = in_sizes[0] / NBINS;
  const int blocks = (n_rays + RAYS_PER_BLOCK - 1) / RAYS_PER_BLOCK;
  hipLaunchKernelGGL(SamplePDF_kernel, dim3(blocks), dim3(256), 0, stream,
                     bins, weights, u, out, n_rays);
}